// PeronaMalik_13331578487066
// MI455X (gfx1250) — compile-verified
//
#include <hip/hip_runtime.h>

// Perona-Malik diffusion step, fully fused single-pass stencil.
// Bandwidth-bound (128 MB min traffic @ 23.3 TB/s ~= 5.5us); WMMA does not
// apply (no contraction). CDNA5-specific path used: async global->LDS tile
// staging (ASYNCcnt) when the builtin exists, LDS two-stage g/h computation.

#define TH 32
#define TW 32
#define XS_W 36   // LDS row stride for x tile (34 cols used)
#define GS_W 34   // LDS row stride for g/h tiles (33 cols used)

#if defined(__HIP_DEVICE_COMPILE__) && \
    __has_builtin(__builtin_amdgcn_global_load_async_to_lds_b32) && \
    __has_builtin(__builtin_amdgcn_s_wait_asynccnt)
#define PM_ASYNC 1
#else
#define PM_ASYNC 0
#endif

typedef __attribute__((address_space(1))) int* pm_global_i32;
typedef __attribute__((address_space(3))) int* pm_lds_i32;

__device__ __forceinline__ int reflect_idx(int t, int n) {
  // jnp.pad 'reflect' with pad 1..2: -1 -> 1, n -> n-2
  t = (t < 0) ? -t : t;
  return (t >= n) ? (2 * n - 2 - t) : t;
}

__global__ __launch_bounds__(256) void pm_fused(const float* __restrict__ x,
                                                float* __restrict__ out,
                                                int H, int W) {
  constexpr float S      = 0.70710678118654752440f; // 1/sqrt(2)
  constexpr float INV_K2 = 1.0f / 900.0f;           // 1/KAPPA^2

  __shared__ float xs[34 * XS_W]; // x tile with 1-halo, reflected at image edge
  __shared__ float gs[33 * GS_W]; // g = fx*c on rows i0..i0+32, cols j0-1..j0+31
  __shared__ float hs[33 * GS_W]; // h = fy*c, same region

  const int j0 = blockIdx.x * TW;
  const int i0 = blockIdx.y * TH;
  const size_t base = (size_t)blockIdx.z * H * W;
  const float* __restrict__ xb = x + base;
  float* __restrict__ ob = out + base;

  const int t = threadIdx.y * 32 + threadIdx.x;

  // ---- Stage A: stage 34x34 x-tile (rows i0-1..i0+32, cols j0-1..j0+32) ----
  for (int e = t; e < 34 * 34; e += 256) {
    const int r  = e / 34;
    const int c  = e - r * 34;
    const int gi = reflect_idx(i0 + r - 1, H);
    const int gj = reflect_idx(j0 + c - 1, W);
#if PM_ASYNC
    __builtin_amdgcn_global_load_async_to_lds_b32(
        (pm_global_i32)(xb + (size_t)gi * W + gj),
        (pm_lds_i32)&xs[r * XS_W + c],
        /*offset=*/0, /*cpol=*/0);
#else
    xs[r * XS_W + c] = xb[(size_t)gi * W + gj];
#endif
  }
#if PM_ASYNC
  __builtin_amdgcn_s_wait_asynccnt(0);
#endif
  __syncthreads();

  // ---- Stage B: g/h over region rows rr=0..32 (i=i0+rr), cols cc=0..32
  //      (j = j0+cc-1). One exp per point. ----
  for (int e = t; e < 33 * 33; e += 256) {
    const int rr = e / 33;
    const int cc = e - rr * 33;
    const float xc = xs[(rr + 1) * XS_W + cc];     // x(i, j)
    const float xe = xs[(rr + 1) * XS_W + cc + 1]; // x(i, j+1)
    const float xn = xs[rr * XS_W + cc];           // x(i-1, j)
    const float fx = S * (xe - xc);
    const float fy = S * (xn - xc);
    const float ci = __expf(-(fx * fx + fy * fy) * INV_K2);
    gs[rr * GS_W + cc] = fx * ci;
    hs[rr * GS_W + cc] = fy * ci;
  }
  __syncthreads();

  // ---- Stage C: divergence with reflect on the g/h fields ----
  const int tx = threadIdx.x;
  for (int k = 0; k < 4; ++k) {
    const int ty = threadIdx.y + 8 * k;
    const int i = i0 + ty;
    const int j = j0 + tx;
    const float gij = gs[ty * GS_W + tx + 1];             // g(i, j)
    const int   cm  = (j == 0) ? 2 : tx;                  // g(i, j-1), reflect
    const float gim = gs[ty * GS_W + cm];
    const float hij = hs[ty * GS_W + tx + 1];             // h(i, j)
    const int   rp  = (i == H - 1) ? (TH - 2) : (ty + 1); // h(i+1, j), reflect
    const float hip = hs[rp * GS_W + tx + 1];
    ob[(size_t)i * W + j] = S * ((gij - gim) + (hij - hip));
  }
}

extern "C" void kernel_launch(void* const* d_in, const int* in_sizes, int n_in,
                              void* d_out, int out_size, void* d_ws, size_t ws_size,
                              hipStream_t stream) {
  (void)n_in; (void)out_size; (void)d_ws; (void)ws_size;
  const float* x = (const float*)d_in[0];
  float* out = (float*)d_out;
  const int H = 1024, W = 1024;
  const int B = in_sizes[0] / (H * W); // 16
  dim3 grid(W / TW, H / TH, B);
  dim3 block(32, 8);
  hipLaunchKernelGGL(pm_fused, grid, block, 0, stream, x, out, H, W);
}